// MicroAdder_61040075210996
// MI455X (gfx1250) — compile-verified
//
#include <hip/hip_runtime.h>
#include <math.h>

typedef __attribute__((ext_vector_type(2))) float v2f;
typedef __attribute__((ext_vector_type(8))) float v8f;

#define TT 34
#define DM 5
#define HD 5
#define NVOCAB 14

__global__ __launch_bounds__(64) void micro_tf_kernel(
    const int*   __restrict__ idx,
    const float* __restrict__ tok_emb,
    const float* __restrict__ s_amp,   const float* __restrict__ s_phase,
    const float* __restrict__ s_slope, const float* __restrict__ s_offset,
    const float* __restrict__ pc_slope, const float* __restrict__ pc_icpt,
    const float* __restrict__ z_hi,    const float* __restrict__ special,
    const float* __restrict__ q_w,     const float* __restrict__ v_w,
    const float* __restrict__ out_A,   const float* __restrict__ out_B,
    const float* __restrict__ q_phase,
    const float* __restrict__ ln1_w,   const float* __restrict__ ln2_w,
    const float* __restrict__ lnf_w,
    const float* __restrict__ fc1_w,   const float* __restrict__ fc1_b,
    const float* __restrict__ fc2_w,   const float* __restrict__ fc2_b,
    const float* __restrict__ head_w,
    float* __restrict__ out)
{
    __shared__ float sK[TT][HD];
    __shared__ float sV[TT][HD];
    __shared__ float sX[TT][HD];
    __shared__ float sW[HD][16];   // fused head_w.T @ tok_emb.T, N padded to 16

    const int tid = threadIdx.x;
    const int b   = blockIdx.x;

    // Precompute fused output projection W[m][n] = sum_j head_w[j,m]*tok_emb[n,j]
    for (int e = tid; e < HD * 16; e += 64) {
        int m = e >> 4, n = e & 15;
        float w = 0.f;
        if (n < NVOCAB)
            w = head_w[0 * DM + m] * tok_emb[n * 2 + 0] +
                head_w[1 * DM + m] * tok_emb[n * 2 + 1];
        sW[m][n] = w;
    }

    float x[DM];
    float qr[HD];

    if (tid < TT) {
        const int t = tid;
        // POS_FLAT mapping (hardcoded from reference)
        int fi;
        if      (t < 10) fi = t;
        else if (t == 10) fi = 11;
        else if (t < 21) fi = t - 11;
        else if (t == 21) fi = 12;
        else if (t < 32) fi = t - 22;
        else if (t == 32) fi = 10;
        else             fi = 13;

        float p0, p1, p2;
        if (fi < 10) {
            float fidx = (float)fi;
            float ang  = 6.283185307179586f * fidx * 0.1f + s_phase[0];
            float corr = 1.f + pc_icpt[0] + pc_slope[0] * fidx;
            p0 = s_amp[0] * cosf(ang) * corr;
            p1 = s_amp[0] * sinf(ang) * corr;
            p2 = (s_slope[0] * fidx + s_offset[0]) * corr;
        } else if (fi == 10) {
            p0 = z_hi[0]; p1 = z_hi[1]; p2 = z_hi[2];
        } else if (fi <= 12) {
            int r2 = fi - 11;
            p0 = special[r2 * 3 + 0]; p1 = special[r2 * 3 + 1]; p2 = special[r2 * 3 + 2];
        } else {
            p0 = p1 = p2 = 0.f;
        }

        int token = idx[b * TT + t];
        x[0] = tok_emb[token * 2 + 0];
        x[1] = tok_emb[token * 2 + 1];
        x[2] = p0; x[3] = p1; x[4] = p2;

        // RMS(ln1)
        float ss = 0.f;
        for (int m = 0; m < DM; m++) ss += x[m] * x[m];
        float rn = rsqrtf(ss * (1.f / DM) + 1e-5f);
        float h[DM];
        for (int m = 0; m < DM; m++) h[m] = x[m] * rn * ln1_w[m];

        // q = h[2:5] @ q_w.T ; v = h[0:2] @ v_w.T ; k == q (pre-rotation)
        float q[HD], v[HD];
        for (int d = 0; d < HD; d++) {
            q[d] = h[2] * q_w[d * 3 + 0] + h[3] * q_w[d * 3 + 1] + h[4] * q_w[d * 3 + 2];
            v[d] = h[0] * v_w[d * 2 + 0] + h[1] * v_w[d * 2 + 1];
        }
        float cph = cosf(q_phase[0]), sph = sinf(q_phase[0]);
        qr[0] = q[0] * cph - q[1] * sph;
        qr[1] = q[0] * sph + q[1] * cph;
        qr[2] = q[2] * cph - q[3] * sph;
        qr[3] = q[2] * sph + q[3] * cph;
        qr[4] = q[4];
        for (int d = 0; d < HD; d++) { sK[t][d] = q[d]; sV[t][d] = v[d]; }
    }
    __syncthreads();

    if (tid < TT) {
        const int t = tid;
        // causal attention row t: online softmax over s<=t
        const float scale = 0.4472135954999579f;  // 1/sqrt(HEAD_DIM)
        float mx = -INFINITY, l = 0.f;
        float acc[HD] = {0.f, 0.f, 0.f, 0.f, 0.f};
        for (int s2 = 0; s2 <= t; s2++) {
            float e = 0.f;
            for (int d = 0; d < HD; d++) e += qr[d] * sK[s2][d];
            e *= scale;
            if (e > mx) {
                float f = __expf(mx - e);
                l *= f;
                for (int d = 0; d < HD; d++) acc[d] *= f;
                mx = e;
            }
            float p = __expf(e - mx);
            l += p;
            for (int d = 0; d < HD; d++) acc[d] += p * sV[s2][d];
        }
        float inv = 1.f / l;
        float o[HD];
        for (int d = 0; d < HD; d++) o[d] = acc[d] * inv;

        // x += (o @ out_A) @ out_B
        float a0 = 0.f, a1 = 0.f;
        for (int d = 0; d < HD; d++) {
            a0 += o[d] * out_A[d * 2 + 0];
            a1 += o[d] * out_A[d * 2 + 1];
        }
        for (int m = 0; m < DM; m++) x[m] += a0 * out_B[0 * DM + m] + a1 * out_B[1 * DM + m];

        // RMS(ln2) + FFN (exact gelu) residual
        float ss = 0.f;
        for (int m = 0; m < DM; m++) ss += x[m] * x[m];
        float rn = rsqrtf(ss * (1.f / DM) + 1e-5f);
        float hh[DM];
        for (int m = 0; m < DM; m++) hh[m] = x[m] * rn * ln2_w[m];
        float f0 = fc1_b[0], f1 = fc1_b[1];
        for (int m = 0; m < DM; m++) {
            f0 += hh[m] * fc1_w[0 * DM + m];
            f1 += hh[m] * fc1_w[1 * DM + m];
        }
        float g0 = 0.5f * f0 * (1.f + erff(f0 * 0.7071067811865476f));
        float g1 = 0.5f * f1 * (1.f + erff(f1 * 0.7071067811865476f));
        for (int m = 0; m < DM; m++)
            x[m] += g0 * fc2_w[m * 2 + 0] + g1 * fc2_w[m * 2 + 1] + fc2_b[m];

        // RMS(lnf) -> stage final x for the WMMA head projection
        ss = 0.f;
        for (int m = 0; m < DM; m++) ss += x[m] * x[m];
        rn = rsqrtf(ss * (1.f / DM) + 1e-5f);
        for (int m = 0; m < DM; m++) sX[t][m] = x[m] * rn * lnf_w[m];
    }
    __syncthreads();

    // ---- logits = sX(34x5) @ sW(5x16) via V_WMMA_F32_16X16X4_F32 ----
    // 3 M-tiles of 16; K=5 split into two K=4 chunks (second zero-padded).
    // wave0 -> tiles {0,2}, wave1 -> tile {1}. Both waves fully active (EXEC=~0).
    const int lane = tid & 31;
    const int wave = tid >> 5;
    const int half = lane >> 4;      // 0: lanes 0-15, 1: lanes 16-31
    const int lrow = lane & 15;
    const int N    = lrow;

    // B operand (4x16 f32): lanes0-15 hold K0(b.x),K1(b.y); lanes16-31 hold K2,K3
    v2f b0, b1;
    b0.x = sW[half ? 2 : 0][N];
    b0.y = sW[half ? 3 : 1][N];
    b1.x = half ? 0.f : sW[4][N];
    b1.y = 0.f;

    const int ntile = (wave == 0) ? 2 : 1;
    for (int it = 0; it < ntile; it++) {
        int tile = (wave == 0) ? (it == 0 ? 0 : 2) : 1;
        int M = tile * 16 + lrow;    // A: same M for both lane halves
        float x0 = 0.f, x1 = 0.f, x2 = 0.f, x3 = 0.f, x4 = 0.f;
        if (M < TT) { x0 = sX[M][0]; x1 = sX[M][1]; x2 = sX[M][2]; x3 = sX[M][3]; x4 = sX[M][4]; }
        v2f a0, a1;
        a0.x = half ? x2 : x0;       // A 16x4: lanes0-15 K0,K1; lanes16-31 K2,K3
        a0.y = half ? x3 : x1;
        a1.x = half ? 0.f : x4;      // K chunk {4,5,6,7}: only K=4 nonzero
        a1.y = 0.f;

        v8f c = {};
        c = __builtin_amdgcn_wmma_f32_16x16x4_f32(false, a0, false, b0, (short)0, c, false, false);
        c = __builtin_amdgcn_wmma_f32_16x16x4_f32(false, a1, false, b1, (short)0, c, false, false);

        // D 16x16 f32: VGPR r -> M = r + 8*half; lane -> N
        for (int r = 0; r < 8; r++) {
            int Mo = tile * 16 + r + (half ? 8 : 0);
            if (Mo < TT && N < NVOCAB)
                out[((size_t)b * TT + Mo) * NVOCAB + N] = c[r];
        }
    }
}

extern "C" void kernel_launch(void* const* d_in, const int* in_sizes, int n_in,
                              void* d_out, int out_size, void* d_ws, size_t ws_size,
                              hipStream_t stream) {
    const int*   idx      = (const int*)  d_in[0];
    const float* tok_emb  = (const float*)d_in[1];
    const float* s_amp    = (const float*)d_in[2];
    const float* s_phase  = (const float*)d_in[3];
    const float* s_slope  = (const float*)d_in[4];
    const float* s_offset = (const float*)d_in[5];
    const float* pc_slope = (const float*)d_in[6];
    const float* pc_icpt  = (const float*)d_in[7];
    const float* z_hi     = (const float*)d_in[8];
    const float* special  = (const float*)d_in[9];
    const float* q_w      = (const float*)d_in[10];
    const float* v_w      = (const float*)d_in[11];
    const float* out_A    = (const float*)d_in[12];
    const float* out_B    = (const float*)d_in[13];
    const float* q_phase  = (const float*)d_in[14];
    const float* ln1_w    = (const float*)d_in[15];
    const float* ln2_w    = (const float*)d_in[16];
    const float* lnf_w    = (const float*)d_in[17];
    const float* fc1_w    = (const float*)d_in[18];
    const float* fc1_b    = (const float*)d_in[19];
    const float* fc2_w    = (const float*)d_in[20];
    const float* fc2_b    = (const float*)d_in[21];
    const float* head_w   = (const float*)d_in[22];
    float* out = (float*)d_out;

    const int B = in_sizes[0] / TT;   // 65536

    micro_tf_kernel<<<B, 64, 0, stream>>>(
        idx, tok_emb, s_amp, s_phase, s_slope, s_offset, pc_slope, pc_icpt,
        z_hi, special, q_w, v_w, out_A, out_B, q_phase,
        ln1_w, ln2_w, lnf_w, fc1_w, fc1_b, fc2_w, fc2_b, head_w, out);
}